// YOLOX2_30331059044415
// MI455X (gfx1250) — compile-verified
//
#include <hip/hip_runtime.h>
#include <hip/hip_bf16.h>
#include <stdint.h>

// ---------------- constants ----------------
#define NUM_IMG   8
#define NUM_CLS   80
#define NUM_LVL   3
#define HBINS     2048
#define CAP       4096      // max candidates kept per (img,lvl) before sort
#define SORT_N    4096      // bitonic size (pow2 >= CAP, >= 3000)
#define TOPK      1000
#define PER_IMG   (NUM_LVL * TOPK)   // 3000
#define CONF_TH   0.05f
#define NMS_TH    0.6f
#define CLS_OFF   4096.0f
#define CHUNK     256       // anchors per block

typedef unsigned u32x4 __attribute__((ext_vector_type(4)));
typedef unsigned u32x8 __attribute__((ext_vector_type(8)));

// ---------------- CDNA5 helpers ----------------
__device__ __forceinline__ void wait_async0() {
#if __has_builtin(__builtin_amdgcn_s_wait_asynccnt)
  __builtin_amdgcn_s_wait_asynccnt(0);
#else
  asm volatile("s_wait_asynccnt 0" ::: "memory");
#endif
}

__device__ __forceinline__ void wait_tensor0() {
#if __has_builtin(__builtin_amdgcn_s_wait_tensorcnt)
  __builtin_amdgcn_s_wait_tensorcnt(0);
#else
  asm volatile("s_wait_tensorcnt 0" ::: "memory");
#endif
}

// GLOBAL_LOAD_ASYNC_TO_LDS_B32, GVS addressing (ASYNCcnt path).
__device__ __forceinline__ void async_f32_to_lds(unsigned lds_byte_off,
                                                 const float* base_uniform,
                                                 unsigned byte_off) {
  unsigned long long base = (unsigned long long)base_uniform;
  asm volatile("global_load_async_to_lds_b32 %0, %1, %2"
               :: "v"(lds_byte_off), "v"(byte_off), "s"(base)
               : "memory");
}

// TENSOR_LOAD_TO_LDS: 2D tile (tile0 x tile1 elements of 4 bytes), row stride
// stride0 elements, destination LDS byte offset lds_off.  D# packing per
// CDNA5 ISA: group0 = {count|flags, lds_addr,
// global_addr[31:0], global_addr[56:32]|type=2}; group1 carries data_size=4B,
// tensor dims, tile dims and dim0 stride.  Groups 2/3 (dims 2..4) are zero.
__device__ __forceinline__ void tdm_load_2d_f32(unsigned lds_off, const void* gptr,
                                                unsigned tile0, unsigned tile1,
                                                unsigned tdim0, unsigned tdim1,
                                                unsigned stride0) {
  unsigned long long ga = (unsigned long long)gptr;
  u32x4 g0;
  g0[0] = 1u;                                            // count=1 (user, valid)
  g0[1] = lds_off;                                       // lds_addr (bytes)
  g0[2] = (unsigned)ga;                                  // global_addr[31:0]
  g0[3] = (unsigned)((ga >> 32) & 0x01FFFFFFull)         // global_addr[56:32]
        | (2u << 30);                                    // type=2 ("image")
  u32x8 g1;
  g1[0] = (2u << 16);                                    // data_size=2 -> 4 bytes
  g1[1] = (tdim0 & 0xFFFFu) << 16;                       // tensor_dim0[15:0]
  g1[2] = ((tdim0 >> 16) & 0xFFFFu)                      // tensor_dim0[31:16]
        | ((tdim1 & 0xFFFFu) << 16);                     // tensor_dim1[15:0]
  g1[3] = ((tdim1 >> 16) & 0xFFFFu)                      // tensor_dim1[31:16]
        | ((tile0 & 0xFFFFu) << 16);                     // tile_dim0
  g1[4] = (tile1 & 0xFFFFu);                             // tile_dim1 ; tile_dim2=0
  g1[5] = stride0;                                       // tensor_dim0_stride[31:0]
  g1[6] = 0u;                                            // stride hi / dim1_stride lo
  g1[7] = 0u;
  u32x4 g2 = {0u, 0u, 0u, 0u};
  u32x4 g3 = {0u, 0u, 0u, 0u};
  asm volatile("tensor_load_to_lds %0, %1, %2, %3"
               :: "s"(g0), "s"(g1), "s"(g2), "s"(g3)
               : "memory");
}

__device__ __forceinline__ float sigm(float x) {
  return 1.0f / (1.0f + __expf(-x));
}

// ---------------- K0: zero workspace control region ----------------
__global__ void k_zero(int* __restrict__ p, int n) {
  int t = blockIdx.x * blockDim.x + threadIdx.x;
  if (t < n) p[t] = 0;
}

// ---------------- K1: per-level score histogram ----------------
// grid = (A/CHUNK, NUM_IMG), block = CHUNK
__global__ void k_hist(const float* __restrict__ obj, const float* __restrict__ cls,
                       int A, int lvl, int* __restrict__ hist) {
  __shared__ float s_cls[NUM_CLS * CHUNK];   // 80 KB tile, TDM-filled
  __shared__ float s_obj[CHUNK];
  __shared__ int   s_hist[HBINS];
  const int tid = threadIdx.x;
  const int chunk = blockIdx.x;
  const int img = blockIdx.y;

  for (int i = tid; i < HBINS; i += blockDim.x) s_hist[i] = 0;

  const float* ctile = cls + (size_t)img * NUM_CLS * A + (size_t)chunk * CHUNK;

  // wave 0 issues one TDM op for the whole 80x256 cls tile (TENSORcnt)
  if (tid < 32) {
    unsigned cls_lds = (unsigned)(uintptr_t)(&s_cls[0]);
    tdm_load_2d_f32(cls_lds, ctile, CHUNK, NUM_CLS,
                    (unsigned)A, (unsigned)NUM_CLS, (unsigned)A);
    wait_tensor0();
  }
  // all threads async-stage their obj element (ASYNCcnt)
  unsigned lds_off = (unsigned)(uintptr_t)(&s_obj[tid]);
  const float* obase = obj + (size_t)img * A + (size_t)chunk * CHUNK;
  async_f32_to_lds(lds_off, obase, tid * 4u);
  // warm L2 with the next chunk's first cls row
  if (chunk + 1 < gridDim.x)
    __builtin_prefetch(ctile + CHUNK, 0, 1);
  wait_async0();
  __syncthreads();

  const float so = sigm(s_obj[tid]);
  for (int c = 0; c < NUM_CLS; ++c) {
    float s = sqrtf(so * sigm(s_cls[c * CHUNK + tid]));
    int bin = (int)(s * HBINS);
    bin = bin < 0 ? 0 : (bin > HBINS - 1 ? HBINS - 1 : bin);
    atomicAdd(&s_hist[bin], 1);
  }
  __syncthreads();
  int* gh = hist + ((size_t)img * NUM_LVL + lvl) * HBINS;
  for (int i = tid; i < HBINS; i += blockDim.x)
    if (s_hist[i]) atomicAdd(&gh[i], s_hist[i]);
}

// ---------------- K2: find cutoff bin per (img,lvl) ----------------
__global__ void k_cutoff(const int* __restrict__ hist, int* __restrict__ cut) {
  int t = threadIdx.x;
  if (t >= NUM_IMG * NUM_LVL) return;
  const int* h = hist + (size_t)t * HBINS;
  int acc = 0, bin = 0;
  for (int b = HBINS - 1; b >= 0; --b) {
    acc += h[b];
    if (acc >= TOPK) { bin = b; break; }
  }
  cut[t] = bin;
}

// ---------------- K3: compact candidates above cutoff ----------------
__global__ void k_compact(const float* __restrict__ obj, const float* __restrict__ cls,
                          int A, int lvl, const int* __restrict__ cut,
                          int* __restrict__ cnt,
                          float* __restrict__ cand_s, int* __restrict__ cand_i) {
  __shared__ float s_cls[NUM_CLS * CHUNK];
  __shared__ float s_obj[CHUNK];
  const int tid = threadIdx.x;
  const int chunk = blockIdx.x;
  const int img = blockIdx.y;
  const int pair = img * NUM_LVL + lvl;
  const int cutoff = cut[pair];

  const float* ctile = cls + (size_t)img * NUM_CLS * A + (size_t)chunk * CHUNK;
  if (tid < 32) {
    unsigned cls_lds = (unsigned)(uintptr_t)(&s_cls[0]);
    tdm_load_2d_f32(cls_lds, ctile, CHUNK, NUM_CLS,
                    (unsigned)A, (unsigned)NUM_CLS, (unsigned)A);
    wait_tensor0();
  }
  unsigned lds_off = (unsigned)(uintptr_t)(&s_obj[tid]);
  const float* obase = obj + (size_t)img * A + (size_t)chunk * CHUNK;
  async_f32_to_lds(lds_off, obase, tid * 4u);
  wait_async0();
  __syncthreads();

  const int a = chunk * CHUNK + tid;
  const float so = sigm(s_obj[tid]);
  float* cs = cand_s + (size_t)pair * CAP;
  int*   ci = cand_i + (size_t)pair * CAP;
  for (int c = 0; c < NUM_CLS; ++c) {
    float s = sqrtf(so * sigm(s_cls[c * CHUNK + tid]));
    int bin = (int)(s * HBINS);
    bin = bin < 0 ? 0 : (bin > HBINS - 1 ? HBINS - 1 : bin);
    if (bin >= cutoff) {
      int pos = atomicAdd(&cnt[pair], 1);
      if (pos < CAP) { cs[pos] = s; ci[pos] = a * NUM_CLS + c; }
    }
  }
}

// ---------------- in-LDS bitonic sort, descending ----------------
__device__ void bitonic_desc(float* k, int* v, int n, int tid, int nt) {
  for (int kk = 2; kk <= n; kk <<= 1) {
    for (int j = kk >> 1; j > 0; j >>= 1) {
      for (int i = tid; i < n; i += nt) {
        int ixj = i ^ j;
        if (ixj > i) {
          float a = k[i], b = k[ixj];
          bool up = ((i & kk) == 0);
          if (up ? (a < b) : (a > b)) {
            k[i] = b; k[ixj] = a;
            int t = v[i]; v[i] = v[ixj]; v[ixj] = t;
          }
        }
      }
      __syncthreads();
    }
  }
}

// ---------------- K4: per-level top-1000 + box decode ----------------
// grid = NUM_IMG, block = 512
__global__ void k_level_final(const float* __restrict__ cand_s, const int* __restrict__ cand_i,
                              const int* __restrict__ cnt, const float* __restrict__ reg,
                              int A, int W, float stride, int lvl,
                              float* __restrict__ cscore, int* __restrict__ clabel,
                              float* __restrict__ cbox, int* __restrict__ cvalid) {
  __shared__ float sk[SORT_N];
  __shared__ int   sv[SORT_N];
  const int img = blockIdx.x, tid = threadIdx.x, nt = blockDim.x;
  const int pair = img * NUM_LVL + lvl;
  int count = cnt[pair]; if (count > CAP) count = CAP;
  const float* cs = cand_s + (size_t)pair * CAP;
  const int*   ci = cand_i + (size_t)pair * CAP;
  for (int i = tid; i < SORT_N; i += nt) {
    if (i < count) { sk[i] = cs[i]; sv[i] = ci[i]; }
    else           { sk[i] = -1e30f; sv[i] = 0; }
  }
  __syncthreads();
  bitonic_desc(sk, sv, SORT_N, tid, nt);

  for (int t = tid; t < TOPK; t += nt) {
    float s = sk[t];
    int idx = sv[t];
    int anchor = idx / NUM_CLS, label = idx % NUM_CLS;
    int x = anchor % W, y = anchor / W;
    const float* rb = reg + (size_t)img * 4 * A + anchor;
    float rx = rb[0], ry = rb[(size_t)A], rw = rb[2 * (size_t)A], rh = rb[3 * (size_t)A];
    float cx = rx * stride + ((float)x + 0.5f) * stride;
    float cy = ry * stride + ((float)y + 0.5f) * stride;
    float w  = __expf(rw) * stride;
    float h  = __expf(rh) * stride;
    int o = img * PER_IMG + lvl * TOPK + t;
    int valid = s > CONF_TH ? 1 : 0;
    cscore[o] = valid ? s : 0.0f;
    clabel[o] = label;
    cvalid[o] = valid;
    cbox[(size_t)o * 4 + 0] = cx - w * 0.5f;
    cbox[(size_t)o * 4 + 1] = cy - h * 0.5f;
    cbox[(size_t)o * 4 + 2] = cx + w * 0.5f;
    cbox[(size_t)o * 4 + 3] = cy + h * 0.5f;
  }
}

// ---------------- K5: per-image descending sort of 3000 ----------------
// grid = NUM_IMG, block = 512
__global__ void k_img_sort(const float* __restrict__ cscore, const int* __restrict__ clabel,
                           const float* __restrict__ cbox, const int* __restrict__ cvalid,
                           float* __restrict__ sscore, int* __restrict__ slabel,
                           float* __restrict__ sbox, int* __restrict__ svalid) {
  __shared__ float sk[SORT_N];
  __shared__ int   sv[SORT_N];
  const int img = blockIdx.x, tid = threadIdx.x, nt = blockDim.x;
  for (int i = tid; i < SORT_N; i += nt) {
    if (i < PER_IMG) { sk[i] = cscore[img * PER_IMG + i]; sv[i] = i; }
    else             { sk[i] = -1e30f; sv[i] = 0; }
  }
  __syncthreads();
  bitonic_desc(sk, sv, SORT_N, tid, nt);
  for (int t = tid; t < PER_IMG; t += nt) {
    int src = sv[t];
    int o = img * PER_IMG + t;
    int s_ = img * PER_IMG + src;
    sscore[o] = cscore[s_];
    slabel[o] = clabel[s_];
    svalid[o] = cvalid[s_];
    for (int q = 0; q < 4; ++q) sbox[(size_t)o * 4 + q] = cbox[(size_t)s_ * 4 + q];
  }
}

// ---------------- K6: greedy NMS (class-offset boxes), per image ----------------
// grid = NUM_IMG, block = 1024. Boxes + keep bitmask in LDS.
#define KM_WORDS ((PER_IMG + 31) / 32)
__global__ void k_nms(const float* __restrict__ sbox, const int* __restrict__ slabel,
                      const int* __restrict__ svalid, int* __restrict__ keep) {
  __shared__ float x1[PER_IMG], y1[PER_IMG], x2[PER_IMG], y2[PER_IMG];
  __shared__ unsigned km[KM_WORDS];
  const int img = blockIdx.x, tid = threadIdx.x, nt = blockDim.x;
  for (int i = tid; i < PER_IMG; i += nt) {
    float off = (float)slabel[img * PER_IMG + i] * CLS_OFF;
    const float* b = sbox + (size_t)(img * PER_IMG + i) * 4;
    x1[i] = b[0] + off; y1[i] = b[1] + off;
    x2[i] = b[2] + off; y2[i] = b[3] + off;
  }
  for (int i = tid; i < KM_WORDS; i += nt) km[i] = 0xFFFFFFFFu;
  __syncthreads();

  for (int i = 0; i < PER_IMG; ++i) {
    bool keep_i = (km[i >> 5] >> (i & 31)) & 1u;
    if (keep_i) {
      float xi1 = x1[i], yi1 = y1[i], xi2 = x2[i], yi2 = y2[i];
      float ai = fmaxf(xi2 - xi1, 0.f) * fmaxf(yi2 - yi1, 0.f);
      for (int j = i + 1 + tid; j < PER_IMG; j += nt) {
        float aj = fmaxf(x2[j] - x1[j], 0.f) * fmaxf(y2[j] - y1[j], 0.f);
        float iw = fmaxf(fminf(xi2, x2[j]) - fmaxf(xi1, x1[j]), 0.f);
        float ih = fmaxf(fminf(yi2, y2[j]) - fmaxf(yi1, y1[j]), 0.f);
        float inter = iw * ih;
        float iou = inter / (ai + aj - inter + 1e-9f);
        if (iou > NMS_TH) atomicAnd(&km[j >> 5], ~(1u << (j & 31)));
      }
    }
    __syncthreads();
  }
  for (int j = tid; j < PER_IMG; j += nt) {
    int k = (((km[j >> 5] >> (j & 31)) & 1u) && svalid[img * PER_IMG + j]) ? 1 : 0;
    keep[img * PER_IMG + j] = k;
  }
}

// ---------------- K7: pack outputs (bboxes, scores, labels, keep) ----------------
#define NB (NUM_IMG * PER_IMG)   // 24000
__global__ void k_output(const float* __restrict__ sscore, const int* __restrict__ slabel,
                         const float* __restrict__ sbox, const int* __restrict__ keep,
                         float* __restrict__ out) {
  int t = blockIdx.x * blockDim.x + threadIdx.x;
  if (t < NB * 4)            out[t] = sbox[t];
  else if (t < NB * 5)       out[t] = sscore[t - NB * 4];
  else if (t < NB * 6)       out[t] = (float)slabel[t - NB * 5];
  else if (t < NB * 7)       out[t] = keep[t - NB * 6] ? 1.0f : 0.0f;
}

// ---------------- host side ----------------
extern "C" void kernel_launch(void* const* d_in, const int* in_sizes, int n_in,
                              void* d_out, int out_size, void* d_ws, size_t ws_size,
                              hipStream_t stream) {
  (void)in_sizes; (void)n_in; (void)out_size; (void)ws_size;
  const float* obj[NUM_LVL] = { (const float*)d_in[0], (const float*)d_in[3], (const float*)d_in[6] };
  const float* cls[NUM_LVL] = { (const float*)d_in[1], (const float*)d_in[4], (const float*)d_in[7] };
  const float* reg[NUM_LVL] = { (const float*)d_in[2], (const float*)d_in[5], (const float*)d_in[8] };
  const int A_[NUM_LVL] = { 128 * 128, 64 * 64, 32 * 32 };
  const int W_[NUM_LVL] = { 128, 64, 32 };
  const float S_[NUM_LVL] = { 8.f, 16.f, 32.f };

  char* ws = (char*)d_ws;
  size_t o = 0;
  auto alloc = [&](size_t bytes) { size_t r = o; o = (o + bytes + 255) & ~(size_t)255; return r; };

  // control region (zeroed each call): hist + cut + cnt, packed contiguously
  size_t o_hist = o; o += (size_t)NUM_IMG * NUM_LVL * HBINS * 4;
  size_t o_cut  = o; o += (size_t)NUM_IMG * NUM_LVL * 4;
  size_t o_cnt  = o; o += (size_t)NUM_IMG * NUM_LVL * 4;
  int n_zero = (int)((o - o_hist) / 4);
  o = (o + 255) & ~(size_t)255;

  size_t o_cand_s = alloc((size_t)NUM_IMG * NUM_LVL * CAP * 4);
  size_t o_cand_i = alloc((size_t)NUM_IMG * NUM_LVL * CAP * 4);
  size_t o_cscore = alloc((size_t)NB * 4);
  size_t o_clabel = alloc((size_t)NB * 4);
  size_t o_cvalid = alloc((size_t)NB * 4);
  size_t o_cbox   = alloc((size_t)NB * 4 * 4);
  size_t o_sscore = alloc((size_t)NB * 4);
  size_t o_slabel = alloc((size_t)NB * 4);
  size_t o_svalid = alloc((size_t)NB * 4);
  size_t o_sbox   = alloc((size_t)NB * 4 * 4);
  size_t o_keep   = alloc((size_t)NB * 4);

  int*   hist   = (int*)(ws + o_hist);
  int*   cut    = (int*)(ws + o_cut);
  int*   cnt    = (int*)(ws + o_cnt);
  float* cand_s = (float*)(ws + o_cand_s);
  int*   cand_i = (int*)(ws + o_cand_i);
  float* cscore = (float*)(ws + o_cscore);
  int*   clabel = (int*)(ws + o_clabel);
  int*   cvalid = (int*)(ws + o_cvalid);
  float* cbox   = (float*)(ws + o_cbox);
  float* sscore = (float*)(ws + o_sscore);
  int*   slabel = (int*)(ws + o_slabel);
  int*   svalid = (int*)(ws + o_svalid);
  float* sbox   = (float*)(ws + o_sbox);
  int*   keep   = (int*)(ws + o_keep);
  float* out    = (float*)d_out;

  k_zero<<<(n_zero + 255) / 256, 256, 0, stream>>>(hist, n_zero);

  for (int l = 0; l < NUM_LVL; ++l)
    k_hist<<<dim3(A_[l] / CHUNK, NUM_IMG), CHUNK, 0, stream>>>(obj[l], cls[l], A_[l], l, hist);

  k_cutoff<<<1, 32, 0, stream>>>(hist, cut);

  for (int l = 0; l < NUM_LVL; ++l)
    k_compact<<<dim3(A_[l] / CHUNK, NUM_IMG), CHUNK, 0, stream>>>(obj[l], cls[l], A_[l], l,
                                                                  cut, cnt, cand_s, cand_i);

  for (int l = 0; l < NUM_LVL; ++l)
    k_level_final<<<NUM_IMG, 512, 0, stream>>>(cand_s, cand_i, cnt, reg[l],
                                               A_[l], W_[l], S_[l], l,
                                               cscore, clabel, cbox, cvalid);

  k_img_sort<<<NUM_IMG, 512, 0, stream>>>(cscore, clabel, cbox, cvalid,
                                          sscore, slabel, sbox, svalid);

  k_nms<<<NUM_IMG, 1024, 0, stream>>>(sbox, slabel, svalid, keep);

  k_output<<<(NB * 7 + 255) / 256, 256, 0, stream>>>(sscore, slabel, sbox, keep, out);
}